// GNN_FULL_CLASS_24275155157633
// MI455X (gfx1250) — compile-verified
//
#include <hip/hip_runtime.h>

// ---------------------------------------------------------------------------
// GNN forward for MI455X (gfx1250, wave32, WMMA).
// Strategy: fp32 WMMA (v_wmma_f32_16x16x4_f32) GEMMs with fused bias+LeakyReLU
// and BN-stat accumulation; BatchNorm folded into the next layer's weights by
// a tiny per-column kernel; edge gathers fused into GEMM A-tile staging;
// unused 'global' MLP skipped entirely.
// ---------------------------------------------------------------------------

typedef __attribute__((ext_vector_type(2))) float v2f;
typedef __attribute__((ext_vector_type(8))) float v8f;

#define BN_EPS   1e-5f
#define LRELU    0.01f
#define BLK_ROWS 128        // rows of X per block (8 waves x 16 rows)
#define NTHREADS 256
#define KC       32         // K-chunk staged in LDS (all K here are multiples of 32)
#define KCPAD    36         // padded K stride (bank-conflict break)
#define NPAD     132        // padded N stride for W tile
#define MAXNO    128        // max output width of any layer

__global__ void zero_kernel(float* __restrict__ p, long n) {
  long i = (long)blockIdx.x * blockDim.x + threadIdx.x;
  if (i < n) p[i] = 0.f;
}

// Y[M,Nout] = act( concat-gather(A0/idx0|A1/idx1|A2) [M,K] @ W[K,Nout] + bias )
// Optionally accumulates per-column sum / sumsq of the activated output into
// stats[0..Nout) and stats[256..256+Nout) via fp32 atomics.
template <int NT>
__global__ void __launch_bounds__(NTHREADS)
gemm_mlp_kernel(const float* __restrict__ A0, const int* __restrict__ idx0, int w0,
                const float* __restrict__ A1, const int* __restrict__ idx1, int w1,
                const float* __restrict__ A2, int w2,
                const float* __restrict__ W,  const float* __restrict__ bias,
                float* __restrict__ Y, int M, int K, int Nout,
                int act, float* __restrict__ stats)
{
  __shared__ float Xs[BLK_ROWS * KCPAD];
  __shared__ float Ws[KC * NPAD];
  __shared__ float Bs[MAXNO];

  const int tid     = threadIdx.x;
  const int wave    = tid >> 5;
  const int lane    = tid & 31;
  const int lh      = lane >> 4;     // lane half (0/1)
  const int ln      = lane & 15;
  const int rowbase = blockIdx.x * BLK_ROWS;
  const int mrow    = rowbase + wave * 16;   // wave-uniform
  const int w01     = w0 + w1;
  constexpr int NP  = NT * 16;               // padded output width we stage

  v8f acc[NT];
  #pragma unroll
  for (int t = 0; t < NT; ++t)
    #pragma unroll
    for (int i = 0; i < 8; ++i) acc[t][i] = 0.f;

  for (int kc = 0; kc < K; kc += KC) {
    __syncthreads();
    // ---- stage A-tile (gathered, possibly 3 concat segments) ----
    for (int i = tid; i < BLK_ROWS * KC; i += NTHREADS) {
      int m    = i >> 5;          // 32 threads per row => coalesced k
      int kk   = i & 31;
      int k    = kc + kk;
      int grow = rowbase + m;
      float v = 0.f;
      if (grow < M) {
        if (k < w0) {
          int r = idx0 ? idx0[grow] : grow;
          v = A0[(long)r * w0 + k];
        } else if (k < w01) {
          int r = idx1 ? idx1[grow] : grow;
          v = A1[(long)r * w1 + (k - w0)];
        } else {
          v = A2[(long)grow * w2 + (k - w01)];
        }
      }
      Xs[m * KCPAD + kk] = v;
    }
    // ---- stage W-chunk (zero-pad cols beyond Nout, e.g. Nout==1) ----
    for (int i = tid; i < KC * NP; i += NTHREADS) {
      int kk = i / NP;            // NP is a power of two -> shift
      int n  = i & (NP - 1);
      Ws[kk * NPAD + n] = (n < Nout) ? W[(long)(kc + kk) * Nout + n] : 0.f;
    }
    if (kc == 0) {
      for (int n = tid; n < MAXNO; n += NTHREADS)
        Bs[n] = (n < Nout) ? bias[n] : 0.f;
    }
    __syncthreads();

    if (mrow < M) {               // wave-uniform branch; EXEC stays full
      #pragma unroll
      for (int k0 = 0; k0 < KC; k0 += 4) {
        // A frag: lanes 0-15 -> K = k0,k0+1 ; lanes 16-31 -> K = k0+2,k0+3
        v2f a;
        const float* xp = &Xs[(wave * 16 + ln) * KCPAD + k0 + 2 * lh];
        a.x = xp[0];
        a.y = xp[1];
        #pragma unroll
        for (int t = 0; t < NT; ++t) {
          v2f b;
          const float* wp = &Ws[(k0 + 2 * lh) * NPAD + t * 16 + ln];
          b.x = wp[0];
          b.y = wp[NPAD];
          acc[t] = __builtin_amdgcn_wmma_f32_16x16x4_f32(
              false, a, false, b, (short)0, acc[t], false, false);
        }
      }
    }
  }

  // ---- epilogue: bias, LeakyReLU, store, BN stats ----
  if (mrow < M) {
    #pragma unroll
    for (int t = 0; t < NT; ++t) {
      int col  = t * 16 + ln;
      float bv = Bs[col];
      float s = 0.f, sq = 0.f;
      #pragma unroll
      for (int i = 0; i < 8; ++i) {
        int m    = i + 8 * lh;                 // C/D layout: VGPR i -> M=i / i+8
        int grow = mrow + m;
        float c  = acc[t][i] + bv;
        if (act) c = (c > 0.f) ? c : LRELU * c;
        if (grow < M && col < Nout) {
          Y[(long)grow * Nout + col] = c;
          s  += c;
          sq += c * c;
        }
      }
      if (stats && col < Nout) {
        atomicAdd(&stats[col], s);
        atomicAdd(&stats[256 + col], sq);
      }
    }
  }
}

// Fold BN(prev-layer stats, g, be) into next layer's weights:
//   Wf[k][n] = scale[k]*Wn[k][n];  bf[n] = bn[n] + sum_k shift[k]*Wn[k][n]
__global__ void bn_fold_kernel(const float* __restrict__ stats, float invM,
                               const float* __restrict__ g, const float* __restrict__ be,
                               const float* __restrict__ Wn, const float* __restrict__ bn,
                               float* __restrict__ Wf, float* __restrict__ bf,
                               int Kd, int Nd)
{
  int n = blockIdx.x;
  int t = threadIdx.x;
  float a = 0.f;
  for (int k = t; k < Kd; k += 64) {
    float mean = stats[k] * invM;
    float var  = stats[256 + k] * invM - mean * mean;
    float sc   = g[k] * rsqrtf(var + BN_EPS);
    float sh   = be[k] - mean * sc;
    float w    = Wn[(long)k * Nd + n];
    Wf[(long)k * Nd + n] = sc * w;
    a += sh * w;
  }
  __shared__ float red[64];
  red[t] = a;
  __syncthreads();
  for (int o = 32; o > 0; o >>= 1) {
    if (t < o) red[t] += red[t + o];
    __syncthreads();
  }
  if (t == 0) bf[n] = bn[n] + red[0];
}

// agg[col[e], c] += msg[e, c]  (128-wide messages)
__global__ void scatter_add_kernel(const float* __restrict__ msg,
                                   const int* __restrict__ col,
                                   float* __restrict__ agg, long total)
{
  long i = (long)blockIdx.x * blockDim.x + threadIdx.x;
  if (i < total) {
    long e = i >> 7;
    int  c = (int)(i & 127);
    atomicAdd(&agg[(long)col[e] * 128 + c], msg[i]);
  }
}

// ---------------------------------------------------------------------------
// Host-side orchestration
// ---------------------------------------------------------------------------
struct Seg { const float* p; const int* idx; int w; };

static void launch_gemm(hipStream_t st, Seg s0, Seg s1, Seg s2,
                        const float* W, const float* b, float* Y,
                        int M, int K, int Nout, int act, float* stats)
{
  dim3 grid((unsigned)((M + BLK_ROWS - 1) / BLK_ROWS)), block(NTHREADS);
  int nt = (Nout + 15) / 16;
#define LNCH(NTV) gemm_mlp_kernel<NTV><<<grid, block, 0, st>>>( \
      s0.p, s0.idx, s0.w, s1.p, s1.idx, s1.w, s2.p, s2.w, W, b, Y, M, K, Nout, act, stats)
  switch (nt) {
    case 1: LNCH(1); break;
    case 2: LNCH(2); break;
    case 4: LNCH(4); break;
    default: LNCH(8); break;
  }
#undef LNCH
}

// P = [w0,b0,g0,be0,w1,b1,g1,be1,w2,b2]; dims din -> h -> h -> dout
static void run_mlp(hipStream_t st, Seg s0, Seg s1, Seg s2,
                    const float* const* P, int M, int din, int h, int dout,
                    float* tmp1, float* tmp2, float* out,
                    float* S, float* WF, float* BF)
{
  float invM = 1.0f / (float)M;
  Seg none{nullptr, nullptr, 0};
  zero_kernel<<<dim3(2), dim3(256), 0, st>>>(S, 512);
  launch_gemm(st, s0, s1, s2, P[0], P[1], tmp1, M, din, h, 1, S);
  bn_fold_kernel<<<dim3((unsigned)h), dim3(64), 0, st>>>(S, invM, P[2], P[3], P[4], P[5], WF, BF, h, h);
  zero_kernel<<<dim3(2), dim3(256), 0, st>>>(S, 512);
  launch_gemm(st, Seg{tmp1, nullptr, h}, none, none, WF, BF, tmp2, M, h, h, 1, S);
  bn_fold_kernel<<<dim3((unsigned)dout), dim3(64), 0, st>>>(S, invM, P[6], P[7], P[8], P[9], WF, BF, h, dout);
  launch_gemm(st, Seg{tmp2, nullptr, h}, none, none, WF, BF, out, M, h, dout, 0, nullptr);
}

extern "C" void kernel_launch(void* const* d_in, const int* in_sizes, int n_in,
                              void* d_out, int out_size, void* d_ws, size_t ws_size,
                              hipStream_t stream)
{
  const float* x  = (const float*)d_in[0];
  const int*   ei = (const int*)d_in[1];     // [2,E] (int32 under JAX default x64-off)
  const float* ea = (const float*)d_in[2];
  // d_in[3] = batch: unused (global-model output is dead code)

  const int N = in_sizes[0] / 64;
  const int E = in_sizes[2] / 32;
  const int* row = ei;
  const int* col = ei + E;

  // params: 7 MLPs x 10 arrays, insertion order:
  // enc_node, enc_edge, edge, node1, node2, global, last
  const float* P[70];
  for (int i = 0; i < 70; ++i) P[i] = (const float*)d_in[4 + i];
  const float* const* enc_node = P + 0;
  const float* const* enc_edge = P + 10;
  const float* const* edge_m   = P + 20;
  const float* const* node1_m  = P + 30;
  const float* const* node2_m  = P + 40;
  const float* const* last_m   = P + 60;

  // workspace layout (floats)
  float* w = (float*)d_ws;
  long off = 0;
  auto alloc = [&](long n) { float* r = w + off; off += n; return r; };
  float* XA = alloc((long)N * 64);    // enc_x ping
  float* XB = alloc((long)N * 64);    // enc_x pong
  float* EA = alloc((long)E * 32);    // enc_ea ping
  float* EB = alloc((long)E * 32);    // enc_ea pong
  float* B1 = alloc((long)E * 128);   // edge-sized scratch
  float* B2 = alloc((long)E * 128);
  float* G1 = alloc((long)N * 128);   // agg
  float* G2 = alloc((long)N * 128);   // node-sized scratch
  float* G3 = alloc((long)N * 128);
  float* S  = alloc(512);             // BN stats: sum[0..256), sumsq[256..512)
  float* WF = alloc(192 * 128);       // folded weights
  float* BF = alloc(128);             // folded bias
  (void)ws_size; (void)n_in; (void)out_size;

  Seg none{nullptr, nullptr, 0};

  // encoders
  run_mlp(stream, Seg{x,  nullptr, 64}, none, none, enc_node, N, 64, 64, 64, G2, G3, XA, S, WF, BF);
  run_mlp(stream, Seg{ea, nullptr, 32}, none, none, enc_edge, E, 32, 32, 32, B1, B2, EA, S, WF, BF);

  float* cx = XA; float* cxn = XB;
  float* ce = EA; float* cen = EB;
  for (int r = 0; r < 2; ++r) {
    // EdgeModel: concat(enc_x[row](64) | enc_x[col](64) | enc_ea(32)) -> 32
    run_mlp(stream, Seg{cx, row, 64}, Seg{cx, col, 64}, Seg{ce, nullptr, 32},
            edge_m, E, 160, 64, 32, B1, B2, cen, S, WF, BF);
    // NodeModel messages: concat(enc_x[row](64) | enc_ea(32)) -> 128
    run_mlp(stream, Seg{cx, row, 64}, Seg{cen, nullptr, 32}, none,
            node1_m, E, 96, 128, 128, B1, B2, B1, S, WF, BF);
    // segment_sum(msg, col) -> agg[N,128]
    long nagg = (long)N * 128;
    zero_kernel<<<dim3((unsigned)((nagg + 255) / 256)), dim3(256), 0, stream>>>(G1, nagg);
    long tot = (long)E * 128;
    scatter_add_kernel<<<dim3((unsigned)((tot + 255) / 256)), dim3(256), 0, stream>>>(B1, col, G1, tot);
    // node2: concat(enc_x(64) | agg(128)) -> 64
    run_mlp(stream, Seg{cx, nullptr, 64}, Seg{G1, nullptr, 128}, none,
            node2_m, N, 192, 128, 64, G2, G3, cxn, S, WF, BF);
    // GlobalModel skipped: its output u is never consumed.
    float* t1 = cx; cx = cxn; cxn = t1;
    float* t2 = ce; ce = cen; cen = t2;
  }

  // final head: 64 -> 128 -> 128 -> 1
  run_mlp(stream, Seg{cx, nullptr, 64}, none, none, last_m, N, 64, 128, 1,
          G2, G3, (float*)d_out, S, WF, BF);
}